// DualGraphSAGE_11390253269041
// MI455X (gfx1250) — compile-verified
//
#include <hip/hip_runtime.h>

#define NNODES 50000
#define NEDGES 800000
#define INDIM  11
#define HID    128

typedef float v2f __attribute__((ext_vector_type(2)));
typedef float v8f __attribute__((ext_vector_type(8)));

// ---------------------------------------------------------------- utilities
__global__ __launch_bounds__(256) void zero_f32(float* __restrict__ p, int n) {
    int i = blockIdx.x * 256 + threadIdx.x;
    if (i < n) p[i] = 0.0f;
}

__global__ __launch_bounds__(256) void degree_count(const long long* __restrict__ dst,
                                                    float* __restrict__ cnt, int E) {
    int e = blockIdx.x * 256 + threadIdx.x;
    if (e < E) unsafeAtomicAdd(&cnt[(int)dst[e]], 1.0f);
}

// generic scatter-sum: one thread per (edge, feature)
__global__ __launch_bounds__(256) void scatter_sum(const float* __restrict__ x,
                                                   const long long* __restrict__ src,
                                                   const long long* __restrict__ dst,
                                                   float* __restrict__ agg, int E, int F) {
    int tid = blockIdx.x * 256 + threadIdx.x;
    if (tid >= E * F) return;
    int e = tid / F, f = tid - e * F;
    int s = (int)src[e], d = (int)dst[e];
    unsafeAtomicAdd(&agg[(size_t)d * F + f], x[(size_t)s * F + f]);
}

// vectorized scatter-sum for F=128: one thread per (edge, 4-float chunk)
__global__ __launch_bounds__(256) void scatter_sum128(const float* __restrict__ x,
                                                      const long long* __restrict__ src,
                                                      const long long* __restrict__ dst,
                                                      float* __restrict__ agg, int E) {
    int tid = blockIdx.x * 256 + threadIdx.x;  // E * 32 threads
    if (tid >= E * 32) return;
    int e = tid >> 5;
    int c = (tid & 31) * 4;
    int s = (int)src[e], d = (int)dst[e];
    const float4 v = *(const float4*)(x + (size_t)s * HID + c);
    float* a = agg + (size_t)d * HID + c;
    unsafeAtomicAdd(a + 0, v.x);
    unsafeAtomicAdd(a + 1, v.y);
    unsafeAtomicAdd(a + 2, v.z);
    unsafeAtomicAdd(a + 3, v.w);
}

__global__ __launch_bounds__(256) void div_count(float* __restrict__ agg,
                                                 const float* __restrict__ cnt, int N, int F) {
    int tid = blockIdx.x * 256 + threadIdx.x;
    if (tid >= N * F) return;
    int row = tid / F;
    agg[tid] /= fmaxf(cnt[row], 1.0f);
}

// ---------------------------------------------------------------- layer 0 (K=11, scalar)
__global__ __launch_bounds__(256) void sage0(const float* __restrict__ agg,
                                             const float* __restrict__ x,
                                             const float* __restrict__ wl,
                                             const float* __restrict__ wr,
                                             const float* __restrict__ bias,
                                             float* __restrict__ h) {
    int tid = blockIdx.x * 256 + threadIdx.x;  // NNODES * HID
    if (tid >= NNODES * HID) return;
    int o = tid & (HID - 1);
    int i = tid >> 7;
    const float* ar = agg + (size_t)i * INDIM;
    const float* xr = x + (size_t)i * INDIM;
    float acc = bias[o];
#pragma unroll
    for (int k = 0; k < INDIM; ++k)
        acc += ar[k] * wl[k * HID + o] + xr[k] * wr[k * HID + o];
    h[tid] = acc;
}

// ---------------------------------------------------------------- WMMA GEMMs (f32, 16x16x4)
// h[m0..m0+15][n0..n0+15] = agg@wl + xin@wr + bias  (K = 128, row-major weights [K][128])
__global__ __launch_bounds__(32) void sage_gemm_wmma(const float* __restrict__ agg,
                                                     const float* __restrict__ xin,
                                                     const float* __restrict__ wl,
                                                     const float* __restrict__ wr,
                                                     const float* __restrict__ bias,
                                                     float* __restrict__ hout) {
    const int lane = threadIdx.x;          // exactly one wave32, EXEC all ones
    const int g = lane >> 4;               // 0: K pair {0,1}, 1: K pair {2,3}
    const int ml = lane & 15;
    const int m0 = blockIdx.x * 16;
    const int n0 = blockIdx.y * 16;

    const float* arow = agg + (size_t)(m0 + ml) * HID;
    const float* xrow = xin + (size_t)(m0 + ml) * HID;

    v8f acc = {};
#pragma unroll
    for (int t = 0; t < HID / 4; ++t) {
        const int k = 4 * t + 2 * g;
        v2f a;  a.x = arow[k];                        a.y = arow[k + 1];
        v2f bl; bl.x = wl[(size_t)k * HID + n0 + ml]; bl.y = wl[(size_t)(k + 1) * HID + n0 + ml];
        acc = __builtin_amdgcn_wmma_f32_16x16x4_f32(false, a, false, bl, (short)0, acc, false, false);
        v2f x2; x2.x = xrow[k];                       x2.y = xrow[k + 1];
        v2f br; br.x = wr[(size_t)k * HID + n0 + ml]; br.y = wr[(size_t)(k + 1) * HID + n0 + ml];
        acc = __builtin_amdgcn_wmma_f32_16x16x4_f32(false, x2, false, br, (short)0, acc, false, false);
    }
    const float bb = bias[n0 + ml];
#pragma unroll
    for (int r = 0; r < 8; ++r) {
        const int row = m0 + r + 8 * g;    // C/D layout: vgpr r, lane n+16g -> D[r+8g][n]
        hout[(size_t)row * HID + n0 + ml] = acc[r] + bb;
    }
}

// out[M][NC] = (relu?) (xin@w + bias), K=128, weights [K][NC]
template <int NC, bool RELU>
__global__ __launch_bounds__(32) void gemm_wmma_single(const float* __restrict__ xin,
                                                       const float* __restrict__ w,
                                                       const float* __restrict__ bias,
                                                       float* __restrict__ out) {
    const int lane = threadIdx.x;
    const int g = lane >> 4;
    const int ml = lane & 15;
    const int m0 = blockIdx.x * 16;
    const int n0 = blockIdx.y * 16;

    const float* xr = xin + (size_t)(m0 + ml) * HID;
    v8f acc = {};
#pragma unroll
    for (int t = 0; t < HID / 4; ++t) {
        const int k = 4 * t + 2 * g;
        v2f a; a.x = xr[k];                         a.y = xr[k + 1];
        v2f b; b.x = w[(size_t)k * NC + n0 + ml];   b.y = w[(size_t)(k + 1) * NC + n0 + ml];
        acc = __builtin_amdgcn_wmma_f32_16x16x4_f32(false, a, false, b, (short)0, acc, false, false);
    }
    const float bb = bias[n0 + ml];
#pragma unroll
    for (int r = 0; r < 8; ++r) {
        float v = acc[r] + bb;
        if (RELU) v = fmaxf(v, 0.0f);
        out[(size_t)(m0 + r + 8 * g) * NC + n0 + ml] = v;
    }
}

// ---------------------------------------------------------------- layernorm + relu + residual
__global__ __launch_bounds__(256) void ln_relu_res(const float* __restrict__ h,
                                                   const float* __restrict__ gw,
                                                   const float* __restrict__ gb,
                                                   const float* __restrict__ xprev,
                                                   float* __restrict__ xout, int residual) {
    const int row = blockIdx.x * 8 + (threadIdx.x >> 5);   // wave32 per row
    const int lane = threadIdx.x & 31;
    const float4 v = ((const float4*)(h + (size_t)row * HID))[lane];

    float s = v.x + v.y + v.z + v.w;
#pragma unroll
    for (int o = 16; o; o >>= 1) s += __shfl_xor(s, o, 32);
    const float mu = s * (1.0f / HID);

    const float dx = v.x - mu, dy = v.y - mu, dz = v.z - mu, dw = v.w - mu;
    float q = dx * dx + dy * dy + dz * dz + dw * dw;
#pragma unroll
    for (int o = 16; o; o >>= 1) q += __shfl_xor(q, o, 32);
    const float rinv = rsqrtf(q * (1.0f / HID) + 1e-5f);

    const int c = lane * 4;
    float4 o4;
    o4.x = fmaxf(dx * rinv * gw[c + 0] + gb[c + 0], 0.0f);
    o4.y = fmaxf(dy * rinv * gw[c + 1] + gb[c + 1], 0.0f);
    o4.z = fmaxf(dz * rinv * gw[c + 2] + gb[c + 2], 0.0f);
    o4.w = fmaxf(dw * rinv * gw[c + 3] + gb[c + 3], 0.0f);
    if (residual) {
        const float4 xp = ((const float4*)(xprev + (size_t)row * HID))[lane];
        o4.x += xp.x; o4.y += xp.y; o4.z += xp.z; o4.w += xp.w;
    }
    ((float4*)(xout + (size_t)row * HID))[lane] = o4;
}

// ---------------------------------------------------------------- classifier output (64 -> 1)
__global__ __launch_bounds__(256) void cls_out(const float* __restrict__ hmid,
                                               const float* __restrict__ w,
                                               const float* __restrict__ b,
                                               float* __restrict__ out) {
    int i = blockIdx.x * 256 + threadIdx.x;
    if (i >= NNODES) return;
    float acc = b[0];
    const float* hr = hmid + (size_t)i * 64;
#pragma unroll
    for (int k = 0; k < 64; ++k) acc += hr[k] * w[k];
    out[i] = acc;
}

// ---------------------------------------------------------------- launch
extern "C" void kernel_launch(void* const* d_in, const int* in_sizes, int n_in,
                              void* d_out, int out_size, void* d_ws, size_t ws_size,
                              hipStream_t stream) {
    (void)in_sizes; (void)n_in; (void)out_size; (void)ws_size;

    const float* x0 = (const float*)d_in[0];
    const long long* ei = (const long long*)d_in[1];
    const long long* src = ei;
    const long long* dst = ei + NEDGES;

    const float* wl0 = (const float*)d_in[2];
    const float* bl0 = (const float*)d_in[3];
    const float* wr0 = (const float*)d_in[4];
    const float* lw0 = (const float*)d_in[5];
    const float* lb0 = (const float*)d_in[6];
    const float* wl1 = (const float*)d_in[7];
    const float* bl1 = (const float*)d_in[8];
    const float* wr1 = (const float*)d_in[9];
    const float* lw1 = (const float*)d_in[10];
    const float* lb1 = (const float*)d_in[11];
    const float* wl2 = (const float*)d_in[12];
    const float* bl2 = (const float*)d_in[13];
    const float* wr2 = (const float*)d_in[14];
    const float* lw2 = (const float*)d_in[15];
    const float* lb2 = (const float*)d_in[16];
    const float* cw0 = (const float*)d_in[17];
    const float* cb0 = (const float*)d_in[18];
    const float* cw1 = (const float*)d_in[19];
    const float* cb1 = (const float*)d_in[20];

    float* cnt = (float*)d_ws;                 // NNODES
    float* B0 = cnt + NNODES;                  // NNODES*HID
    float* B1 = B0 + (size_t)NNODES * HID;     // NNODES*HID
    float* B2 = B1 + (size_t)NNODES * HID;     // NNODES*HID

    const dim3 gemm_grid(NNODES / 16, HID / 16);
    const dim3 cls_grid(NNODES / 16, 64 / 16);

    // degrees (shared across layers)
    zero_f32<<<(NNODES + 255) / 256, 256, 0, stream>>>(cnt, NNODES);
    degree_count<<<(NEDGES + 255) / 256, 256, 0, stream>>>(dst, cnt, NEDGES);

    // ---- layer 0 (IN_DIM=11): agg->B0, h->B1, x1->B2
    zero_f32<<<(NNODES * INDIM + 255) / 256, 256, 0, stream>>>(B0, NNODES * INDIM);
    scatter_sum<<<(NEDGES * INDIM + 255) / 256, 256, 0, stream>>>(x0, src, dst, B0, NEDGES, INDIM);
    div_count<<<(NNODES * INDIM + 255) / 256, 256, 0, stream>>>(B0, cnt, NNODES, INDIM);
    sage0<<<(NNODES * HID + 255) / 256, 256, 0, stream>>>(B0, x0, wl0, wr0, bl0, B1);
    ln_relu_res<<<NNODES / 8, 256, 0, stream>>>(B1, lw0, lb0, nullptr, B2, 0);

    // ---- layer 1: x_prev=B2, agg->B0, h->B1, x2->B0
    zero_f32<<<(NNODES * HID + 255) / 256, 256, 0, stream>>>(B0, NNODES * HID);
    scatter_sum128<<<(NEDGES * 32 + 255) / 256, 256, 0, stream>>>(B2, src, dst, B0, NEDGES);
    div_count<<<(NNODES * HID + 255) / 256, 256, 0, stream>>>(B0, cnt, NNODES, HID);
    sage_gemm_wmma<<<gemm_grid, 32, 0, stream>>>(B0, B2, wl1, wr1, bl1, B1);
    ln_relu_res<<<NNODES / 8, 256, 0, stream>>>(B1, lw1, lb1, B2, B0, 1);

    // ---- layer 2: x_prev=B0, agg->B1, h->B2, x3->B1
    zero_f32<<<(NNODES * HID + 255) / 256, 256, 0, stream>>>(B1, NNODES * HID);
    scatter_sum128<<<(NEDGES * 32 + 255) / 256, 256, 0, stream>>>(B0, src, dst, B1, NEDGES);
    div_count<<<(NNODES * HID + 255) / 256, 256, 0, stream>>>(B1, cnt, NNODES, HID);
    sage_gemm_wmma<<<gemm_grid, 32, 0, stream>>>(B1, B0, wl2, wr2, bl2, B2);
    ln_relu_res<<<NNODES / 8, 256, 0, stream>>>(B2, lw2, lb2, B0, B1, 1);

    // ---- classifier: hmid = relu(x3@c_w0 + c_b0) -> B0, out = hmid@c_w1 + c_b1
    gemm_wmma_single<64, true><<<cls_grid, 32, 0, stream>>>(B1, cw0, cb0, B0);
    cls_out<<<(NNODES + 255) / 256, 256, 0, stream>>>(B0, cw1, cb1, (float*)d_out);
}